// Self_Attn_1305670058655
// MI455X (gfx1250) — compile-verified
//
#include <hip/hip_runtime.h>
#include <math.h>

typedef __attribute__((ext_vector_type(2))) float v2f;
typedef __attribute__((ext_vector_type(8))) float v8f;

#define N_PIX 9216       // 96*96
#define C_IN  256
#define C_QK  32
#define NTHR  256
#define NTILE (N_PIX / 16)   // 576 16-wide tiles along N

// ---------------------------------------------------------------------------
// K1: q = Wq@style + bq, k = Wk@x + bk, mean-center over channels,
//     store q_hat/k_hat [32][N] and per-block partial squared norms.
// One thread per spatial column n; weights are uniform -> scalar loads.
// ---------------------------------------------------------------------------
__global__ void qk_kernel(const float* __restrict__ x, const float* __restrict__ sf,
                          const float* __restrict__ Wq, const float* __restrict__ bq,
                          const float* __restrict__ Wk, const float* __restrict__ bk,
                          float* __restrict__ qhat, float* __restrict__ khat,
                          float* __restrict__ qpart, float* __restrict__ kpart) {
    const int n = blockIdx.x * blockDim.x + threadIdx.x;
    float accq[C_QK], acck[C_QK];
#pragma unroll
    for (int o = 0; o < C_QK; ++o) { accq[o] = 0.f; acck[o] = 0.f; }

    for (int c = 0; c < C_IN; ++c) {
        const float s  = sf[c * N_PIX + n];   // coalesced
        const float xv = x [c * N_PIX + n];   // coalesced
#pragma unroll
        for (int o = 0; o < C_QK; ++o) {
            accq[o] = fmaf(Wq[o * C_IN + c], s,  accq[o]);  // uniform -> s_load
            acck[o] = fmaf(Wk[o * C_IN + c], xv, acck[o]);
        }
    }
    float sumq = 0.f, sumk = 0.f;
#pragma unroll
    for (int o = 0; o < C_QK; ++o) {
        accq[o] += bq[o]; acck[o] += bk[o];
        sumq += accq[o];  sumk += acck[o];
    }
    const float mq = sumq * (1.f / C_QK);
    const float mk = sumk * (1.f / C_QK);
    float nq = 0.f, nk = 0.f;
#pragma unroll
    for (int o = 0; o < C_QK; ++o) {
        const float qh = accq[o] - mq;
        const float kh = acck[o] - mk;
        qhat[o * N_PIX + n] = qh;   // coalesced
        khat[o * N_PIX + n] = kh;
        nq = fmaf(qh, qh, nq);
        nk = fmaf(kh, kh, nk);
    }
    __shared__ float red[NTHR];
    red[threadIdx.x] = nq; __syncthreads();
    for (int s = NTHR / 2; s > 0; s >>= 1) {
        if (threadIdx.x < s) red[threadIdx.x] += red[threadIdx.x + s];
        __syncthreads();
    }
    if (threadIdx.x == 0) qpart[blockIdx.x] = red[0];
    __syncthreads();
    red[threadIdx.x] = nk; __syncthreads();
    for (int s = NTHR / 2; s > 0; s >>= 1) {
        if (threadIdx.x < s) red[threadIdx.x] += red[threadIdx.x + s];
        __syncthreads();
    }
    if (threadIdx.x == 0) kpart[blockIdx.x] = red[0];
}

// ---------------------------------------------------------------------------
// K2: vT[m][c] = sum_cc Wv[c][cc] * x[cc][m] + bv[c]   (m-major for GEMM-B)
// One block per column m; x column staged through LDS.
// ---------------------------------------------------------------------------
__global__ void v_kernel(const float* __restrict__ x, const float* __restrict__ Wv,
                         const float* __restrict__ bv, float* __restrict__ vT) {
    const int m = blockIdx.x;
    __shared__ float xcol[C_IN];
    xcol[threadIdx.x] = x[threadIdx.x * N_PIX + m];
    __syncthreads();
    const int c = threadIdx.x;
    float acc = bv[c];
#pragma unroll 8
    for (int cc = 0; cc < C_IN; ++cc)
        acc = fmaf(Wv[c * C_IN + cc], xcol[cc], acc);
    vT[m * C_IN + c] = acc;   // coalesced
}

// ---------------------------------------------------------------------------
// K3: invDenom = 1 / (||q_hat||_F * ||k_hat||_F)  (deterministic serial sum)
// ---------------------------------------------------------------------------
__global__ void finalize_kernel(const float* __restrict__ qpart,
                                const float* __restrict__ kpart,
                                int nparts, float* __restrict__ invDenom) {
    float sq = 0.f, sk = 0.f;
    for (int i = 0; i < nparts; ++i) { sq += qpart[i]; sk += kpart[i]; }
    *invDenom = 1.0f / (sqrtf(sq) * sqrtf(sk));
}

// ---------------------------------------------------------------------------
// K4: energy[n][m] = (q_hat^T k_hat)[n][m] * invDenom, one 16x16 tile/wave.
// A[n][k] = qhat[k][n], B[k][m] = khat[k][m]; both tile loads coalesced.
// 8 x V_WMMA_F32_16X16X4_F32 per tile (K = 32).
// ---------------------------------------------------------------------------
__global__ void energy_kernel(const float* __restrict__ qhat,
                              const float* __restrict__ khat,
                              const float* __restrict__ invDenomPtr,
                              float* __restrict__ attn) {
    const int wave = (blockIdx.x * blockDim.x + threadIdx.x) >> 5;
    const int ti = wave / NTILE;          // n tile
    const int tj = wave - ti * NTILE;     // m tile
    const int lane = threadIdx.x & 31;
    const int half = lane >> 4;           // 0: K/M 0..; 1: K+2 / M+8
    const int l16  = lane & 15;
    const int n0 = ti * 16, m0 = tj * 16;

    v8f acc = {};
#pragma unroll
    for (int kk = 0; kk < C_QK; kk += 4) {
        const int kA = kk + 2 * half;
        v2f a, b;
        a.x = qhat[kA * N_PIX + n0 + l16];        // A[n][kA]
        a.y = qhat[(kA + 1) * N_PIX + n0 + l16];  // A[n][kA+1]
        b.x = khat[kA * N_PIX + m0 + l16];        // B[kA][m]
        b.y = khat[(kA + 1) * N_PIX + m0 + l16];
        acc = __builtin_amdgcn_wmma_f32_16x16x4_f32(
            false, a, false, b, (short)0, acc, false, false);
    }
    const float inv = *invDenomPtr;
#pragma unroll
    for (int i = 0; i < 8; ++i) {
        const size_t row = (size_t)(n0 + i + 8 * half);
        attn[row * N_PIX + m0 + l16] = acc[i] * inv;   // coalesced per row
    }
}

// ---------------------------------------------------------------------------
// K5: row-wise softmax in place. One block per row; values held in registers
// so the 340MB matrix is read once and written once.
// ---------------------------------------------------------------------------
__global__ void softmax_kernel(float* __restrict__ attn) {
    const size_t row = blockIdx.x;
    float* p = attn + row * N_PIX;
    const int t = threadIdx.x;
    const int PER = N_PIX / NTHR;   // 36
    float ev[PER];
    float mx = -3.402823466e38f;
#pragma unroll
    for (int j = 0; j < PER; ++j) {
        ev[j] = p[t + j * NTHR];
        mx = fmaxf(mx, ev[j]);
    }
    __shared__ float red[NTHR];
    red[t] = mx; __syncthreads();
    for (int s = NTHR / 2; s > 0; s >>= 1) {
        if (t < s) red[t] = fmaxf(red[t], red[t + s]);
        __syncthreads();
    }
    const float rmax = red[0];
    __syncthreads();
    float sum = 0.f;
#pragma unroll
    for (int j = 0; j < PER; ++j) {
        ev[j] = __expf(ev[j] - rmax);
        sum += ev[j];
    }
    red[t] = sum; __syncthreads();
    for (int s = NTHR / 2; s > 0; s >>= 1) {
        if (t < s) red[t] += red[t + s];
        __syncthreads();
    }
    const float inv = 1.0f / red[0];
#pragma unroll
    for (int j = 0; j < PER; ++j) p[t + j * NTHR] = ev[j] * inv;
}

// ---------------------------------------------------------------------------
// K6: out^T[n][c] = sum_m attn[n][m] * vT[m][c];  out = gamma*out + x.
// One 16x16 tile per wave, K = 9216 via f32 WMMA, 2 accumulators for ILP.
// Wave order groups all 16 c-tiles of an n-tile -> attn rows hit L2.
// vT (9.4MB) is L2-resident on the 192MB L2.
// ---------------------------------------------------------------------------
__global__ void out_kernel(const float* __restrict__ attn,
                           const float* __restrict__ vT,
                           const float* __restrict__ x,
                           const float* __restrict__ gammaPtr,
                           float* __restrict__ out) {
    const int wave = (blockIdx.x * blockDim.x + threadIdx.x) >> 5;
    const int ct = wave & 15;     // c tile (fast -> L2 reuse of attn rows)
    const int nt = wave >> 4;     // n tile
    const int lane = threadIdx.x & 31;
    const int half = lane >> 4;
    const int l16  = lane & 15;
    const int n0 = nt * 16, c0 = ct * 16;

    const float* arow = attn + (size_t)(n0 + l16) * N_PIX + 2 * half;
    const float* bcol = vT + (2 * half) * C_IN + c0 + l16;

    v8f acc0 = {};
    v8f acc1 = {};
    for (int m = 0; m < N_PIX; m += 8) {
        v2f a, b;
        a.x = arow[m];                      // contiguous pair per lane
        a.y = arow[m + 1];
        b.x = bcol[(size_t)m * C_IN];       // coalesced across lanes
        b.y = bcol[(size_t)m * C_IN + C_IN];
        acc0 = __builtin_amdgcn_wmma_f32_16x16x4_f32(
            false, a, false, b, (short)0, acc0, false, false);
        a.x = arow[m + 4];
        a.y = arow[m + 5];
        b.x = bcol[(size_t)(m + 4) * C_IN];
        b.y = bcol[(size_t)(m + 4) * C_IN + C_IN];
        acc1 = __builtin_amdgcn_wmma_f32_16x16x4_f32(
            false, a, false, b, (short)0, acc1, false, false);
    }
    const float g = *gammaPtr;
#pragma unroll
    for (int i = 0; i < 8; ++i) {
        const size_t idx = (size_t)(c0 + l16) * N_PIX + n0 + i + 8 * half;
        out[idx] = fmaf(g, acc0[i] + acc1[i], x[idx]);
    }
}

// ---------------------------------------------------------------------------
extern "C" void kernel_launch(void* const* d_in, const int* in_sizes, int n_in,
                              void* d_out, int out_size, void* d_ws, size_t ws_size,
                              hipStream_t stream) {
    (void)in_sizes; (void)n_in; (void)out_size; (void)ws_size;
    const float* x     = (const float*)d_in[0];
    const float* sf    = (const float*)d_in[1];
    const float* Wq    = (const float*)d_in[2];
    const float* bq    = (const float*)d_in[3];
    const float* Wk    = (const float*)d_in[4];
    const float* bk    = (const float*)d_in[5];
    const float* Wv    = (const float*)d_in[6];
    const float* bv    = (const float*)d_in[7];
    const float* gamma = (const float*)d_in[8];

    float* outPtr  = (float*)d_out;                       // [256][9216]
    float* attnPtr = (float*)d_out + (size_t)C_IN * N_PIX; // [9216][9216]

    // workspace layout (floats)
    float* ws   = (float*)d_ws;
    float* qhat = ws;                                   // 32*9216
    float* khat = qhat + (size_t)C_QK * N_PIX;          // 32*9216
    float* vT   = khat + (size_t)C_QK * N_PIX;          // 9216*256
    float* qpart = vT + (size_t)N_PIX * C_IN;           // 36 (pad 64)
    float* kpart = qpart + 64;                          // 36 (pad 64)
    float* invDenom = kpart + 64;                       // 1

    const int nblk_qk = N_PIX / NTHR;                   // 36
    qk_kernel<<<nblk_qk, NTHR, 0, stream>>>(x, sf, Wq, bq, Wk, bk,
                                            qhat, khat, qpart, kpart);
    v_kernel<<<N_PIX, NTHR, 0, stream>>>(x, Wv, bv, vT);
    finalize_kernel<<<1, 1, 0, stream>>>(qpart, kpart, nblk_qk, invDenom);

    const int ntiles_e = NTILE * NTILE;                 // 331776 waves
    energy_kernel<<<ntiles_e / 8, NTHR, 0, stream>>>(qhat, khat, invDenom, attnPtr);

    softmax_kernel<<<N_PIX, NTHR, 0, stream>>>(attnPtr);

    const int nwaves_o = NTILE * (C_IN / 16);           // 9216 waves
    out_kernel<<<nwaves_o / 8, NTHR, 0, stream>>>(attnPtr, vT, x, gamma, outPtr);
}